// TorchSumLayer_78262894068505
// MI455X (gfx1250) — compile-verified
//
#include <hip/hip_runtime.h>

// Problem constants (from reference)
#define BATCH    8192
#define N_IN     4096
#define N_NODES  2048
#define FAN_IN   8

#define ROWS_PER_BLOCK 16     // 16 rows * 16KB = 256KB LDS per block
#define QROWS          4      // rows per TDM quarter-tile (64KB)
#define BLOCK_THREADS  1024

// ---------------------------------------------------------------------------
// TDM availability / arity handling
// ---------------------------------------------------------------------------
#if defined(__has_builtin)
#  if __has_builtin(__builtin_amdgcn_tensor_load_to_lds) && __has_builtin(__builtin_amdgcn_s_wait_tensorcnt)
#    define HAVE_TDM 1
#  endif
#endif
#ifndef HAVE_TDM
#  define HAVE_TDM 0
#endif

typedef unsigned int v4u __attribute__((ext_vector_type(4)));
typedef int          v8i __attribute__((ext_vector_type(8)));
typedef int          v4i __attribute__((ext_vector_type(4)));

#if HAVE_TDM
// Issue one TDM 2D tile load: `rows` rows of `width_elems` f32, row stride
// `row_stride_elems` f32, from global `gaddr` into LDS byte offset `lds_off`.
// Descriptor bitfields per CDNA5 ISA ch.8 (D# groups 0..3).
__device__ __forceinline__ void tdm_load_2d_f32(const void* gaddr,
                                                unsigned lds_off,
                                                unsigned width_elems,
                                                unsigned rows,
                                                unsigned row_stride_elems) {
  unsigned long long ga = (unsigned long long)gaddr;

  v4u g0;
  g0.x = 1u;                                        // count=1, user desc, no gather
  g0.y = lds_off;                                   // lds_addr (bytes)
  g0.z = (unsigned)(ga & 0xFFFFFFFFu);              // global_addr[31:0]
  g0.w = (unsigned)((ga >> 32) & 0x01FFFFFFu)       // global_addr[56:32]
       | (2u << 30);                                // type = 2 ("image")

  v8i g1;
  g1[0] = (int)(2u << 16);                          // data_size=2 (4 bytes); mask=0
  g1[1] = (int)((width_elems & 0xFFFFu) << 16);     // tensor_dim0[15:0]
  g1[2] = (int)(((width_elems >> 16) & 0xFFFFu)     // tensor_dim0[31:16]
       | ((rows & 0xFFFFu) << 16));                 // tensor_dim1[15:0]
  g1[3] = (int)(((rows >> 16) & 0xFFFFu)            // tensor_dim1[31:16]
       | ((width_elems & 0xFFFFu) << 16));          // tile_dim0
  g1[4] = (int)(rows & 0xFFFFu);                    // tile_dim1 (tile_dim2=0)
  g1[5] = (int)row_stride_elems;                    // tensor_dim0_stride[31:0]
  g1[6] = 0;                                        // stride hi / dim1_stride lo
  g1[7] = 0;
  v4i g2 = {0, 0, 0, 0};                            // 2D: dims 2/3 unused
  v4i g3 = {0, 0, 0, 0};
#if __clang_major__ >= 23
  v8i g4 = {0, 0, 0, 0, 0, 0, 0, 0};
  __builtin_amdgcn_tensor_load_to_lds(g0, g1, g2, g3, g4, 0);
#else
  __builtin_amdgcn_tensor_load_to_lds(g0, g1, g2, g3, 0);
#endif
}
#endif // HAVE_TDM

// ---------------------------------------------------------------------------
// Per-node weighted dot over K=8 gathered pre-exponentiated LDS values:
//   out = log( sum_k pw[k] * E[o[k]] )
// ---------------------------------------------------------------------------
__device__ __forceinline__ float node_logdot(const float* __restrict__ xr,
                                             const float* __restrict__ w,
                                             const int* __restrict__ o) {
  float s0 = 0.f, s1 = 0.f;
#pragma unroll
  for (int k = 0; k < FAN_IN; k += 2) {
    s0 = fmaf(w[k],     xr[o[k]],     s0);   // ds_load_b32 gather + v_fma
    s1 = fmaf(w[k + 1], xr[o[k + 1]], s1);
  }
  return __logf(s0 + s1);
}

// Compute the QROWS rows of quarter Qq (already exp-transformed in LDS)
__device__ __forceinline__ void compute_quarter(int q, const float* __restrict__ xtile,
                                                const float* __restrict__ w0, const int* __restrict__ o0,
                                                const float* __restrict__ w1, const int* __restrict__ o1,
                                                float* __restrict__ out, int rowStart, int n0, int n1) {
  for (int rr = 0; rr < QROWS; ++rr) {
    const int r = q * QROWS + rr;
    const float* xr = xtile + (size_t)r * N_IN;
    float va = node_logdot(xr, w0, o0);
    float vb = node_logdot(xr, w1, o1);
    size_t row = (size_t)(rowStart + r);
    out[row * N_NODES + n0] = va;   // coalesced b32 stores
    out[row * N_NODES + n1] = vb;
  }
}

#if HAVE_TDM
// One pipeline stage: wait for quarter Q's DMA, exp-transform it in place,
// then compute its 4 rows. Q is a template param so the wait count is an ICE.
template <int Q>
__device__ __forceinline__ void do_quarter(float* __restrict__ xtile, int tid,
                                           const float* __restrict__ w0, const int* __restrict__ o0,
                                           const float* __restrict__ w1, const int* __restrict__ o1,
                                           float* __restrict__ out, int rowStart, int n0, int n1) {
  if (tid < 32) __builtin_amdgcn_s_wait_tensorcnt(3 - Q);
  __syncthreads();
  // In-place exp of this quarter: QROWS*N_IN = 16384 floats / 1024 thr = 4x float4
  {
    float4* t4 = reinterpret_cast<float4*>(xtile) + Q * (QROWS * N_IN / 4);
#pragma unroll
    for (int j = 0; j < (QROWS * N_IN / 4) / BLOCK_THREADS; ++j) {
      float4 v = t4[tid + j * BLOCK_THREADS];      // ds_load_b128
      v.x = __expf(v.x); v.y = __expf(v.y);
      v.z = __expf(v.z); v.w = __expf(v.w);
      t4[tid + j * BLOCK_THREADS] = v;             // ds_store_b128
    }
  }
  __syncthreads();
  compute_quarter(Q, xtile, w0, o0, w1, o1, out, rowStart, n0, n1);
}
#endif

// ---------------------------------------------------------------------------
// Main: block = 16 batch rows staged to LDS via 4 pipelined TDM descriptors;
// tile exponentiated in place once (each element reused ~4x on average), then per-node
// linear-domain softmax dot + single v_log. Thread t owns nodes t and t+1024.
// ---------------------------------------------------------------------------
__global__ __launch_bounds__(BLOCK_THREADS)
void spn_sum_kernel(const float* __restrict__ x,
                    const float* __restrict__ pw,
                    const int* __restrict__ idx,
                    float* __restrict__ out) {
  extern __shared__ float xtile[];   // ROWS_PER_BLOCK * N_IN floats (256 KB)
  const int tid = threadIdx.x;
  const int rowStart = blockIdx.x * ROWS_PER_BLOCK;

#if HAVE_TDM
  if (tid < 32) {  // one wave issues all four DMA descriptors
    tdm_load_2d_f32(x + (size_t)(rowStart + 0 * QROWS) * N_IN,
                    0u * QROWS * N_IN * 4u, N_IN, QROWS, N_IN);
    tdm_load_2d_f32(x + (size_t)(rowStart + 1 * QROWS) * N_IN,
                    1u * QROWS * N_IN * 4u, N_IN, QROWS, N_IN);
    tdm_load_2d_f32(x + (size_t)(rowStart + 2 * QROWS) * N_IN,
                    2u * QROWS * N_IN * 4u, N_IN, QROWS, N_IN);
    tdm_load_2d_f32(x + (size_t)(rowStart + 3 * QROWS) * N_IN,
                    3u * QROWS * N_IN * 4u, N_IN, QROWS, N_IN);
  }
#else
  {
    const float4* src = reinterpret_cast<const float4*>(x + (size_t)rowStart * N_IN);
    float4* dst = reinterpret_cast<float4*>(xtile);
    const int total = ROWS_PER_BLOCK * N_IN / 4;
    for (int i = tid; i < total; i += BLOCK_THREADS) {
      float4 v = src[i];
      v.x = __expf(v.x); v.y = __expf(v.y);
      v.z = __expf(v.z); v.w = __expf(v.w);
      dst[i] = v;
    }
  }
#endif

  // Node parameters into registers (L2-resident tables, b128 loads)
  const int n0 = tid;
  const int n1 = tid + BLOCK_THREADS;
  float w0[FAN_IN], w1[FAN_IN];
  int   o0[FAN_IN], o1[FAN_IN];
  {
    const float4* pw0 = reinterpret_cast<const float4*>(pw) + (size_t)n0 * 2;
    const float4* pw1 = reinterpret_cast<const float4*>(pw) + (size_t)n1 * 2;
    const int4*   pi0 = reinterpret_cast<const int4*>(idx) + (size_t)n0 * 2;
    const int4*   pi1 = reinterpret_cast<const int4*>(idx) + (size_t)n1 * 2;
    float4 a = pw0[0], b = pw0[1];
    w0[0]=a.x; w0[1]=a.y; w0[2]=a.z; w0[3]=a.w; w0[4]=b.x; w0[5]=b.y; w0[6]=b.z; w0[7]=b.w;
    a = pw1[0]; b = pw1[1];
    w1[0]=a.x; w1[1]=a.y; w1[2]=a.z; w1[3]=a.w; w1[4]=b.x; w1[5]=b.y; w1[6]=b.z; w1[7]=b.w;
    int4 c = pi0[0], d = pi0[1];
    o0[0]=c.x; o0[1]=c.y; o0[2]=c.z; o0[3]=c.w; o0[4]=d.x; o0[5]=d.y; o0[6]=d.z; o0[7]=d.w;
    c = pi1[0]; d = pi1[1];
    o1[0]=c.x; o1[1]=c.y; o1[2]=c.z; o1[3]=c.w; o1[4]=d.x; o1[5]=d.y; o1[6]=d.z; o1[7]=d.w;
  }

#if HAVE_TDM
  do_quarter<0>(xtile, tid, w0, o0, w1, o1, out, rowStart, n0, n1);
  do_quarter<1>(xtile, tid, w0, o0, w1, o1, out, rowStart, n0, n1);
  do_quarter<2>(xtile, tid, w0, o0, w1, o1, out, rowStart, n0, n1);
  do_quarter<3>(xtile, tid, w0, o0, w1, o1, out, rowStart, n0, n1);
#else
  __syncthreads();
  for (int q = 0; q < ROWS_PER_BLOCK / QROWS; ++q)
    compute_quarter(q, xtile, w0, o0, w1, o1, out, rowStart, n0, n1);
#endif
}

// ---------------------------------------------------------------------------
// Prep: pw[n,k] = softmax(weights[n,:])[k]  (linear domain; 2048 nodes)
// ---------------------------------------------------------------------------
__global__ void spn_prep_kernel(const float* __restrict__ w,
                                float* __restrict__ pw) {
  int n = blockIdx.x * blockDim.x + threadIdx.x;
  if (n >= N_NODES) return;
  const float4* p = reinterpret_cast<const float4*>(w) + (size_t)n * 2;
  float4 a = p[0], b = p[1];
  float v[FAN_IN] = {a.x, a.y, a.z, a.w, b.x, b.y, b.z, b.w};
  float m = v[0];
#pragma unroll
  for (int k = 1; k < FAN_IN; ++k) m = fmaxf(m, v[k]);
  float e[FAN_IN];
  float s = 0.f;
#pragma unroll
  for (int k = 0; k < FAN_IN; ++k) { e[k] = __expf(v[k] - m); s += e[k]; }
  float inv = 1.0f / s;                        // IEEE divide (prep only)
  float4 oa = {e[0] * inv, e[1] * inv, e[2] * inv, e[3] * inv};
  float4 ob = {e[4] * inv, e[5] * inv, e[6] * inv, e[7] * inv};
  float4* q = reinterpret_cast<float4*>(pw) + (size_t)n * 2;
  q[0] = oa;
  q[1] = ob;
}

// ---------------------------------------------------------------------------
// Host entry
// ---------------------------------------------------------------------------
extern "C" void kernel_launch(void* const* d_in, const int* in_sizes, int n_in,
                              void* d_out, int out_size, void* d_ws, size_t ws_size,
                              hipStream_t stream) {
  const float* x   = (const float*)d_in[0];   // [BATCH, N_IN]
  const float* w   = (const float*)d_in[1];   // [N_NODES, FAN_IN]
  const int*   idx = (const int*)d_in[2];     // [N_NODES, FAN_IN]
  float* out = (float*)d_out;                 // [BATCH, N_NODES]
  float* pw  = (float*)d_ws;                  // N_NODES*FAN_IN floats (64 KB)

  spn_prep_kernel<<<(N_NODES + 255) / 256, 256, 0, stream>>>(w, pw);

  const size_t lds_bytes = (size_t)ROWS_PER_BLOCK * N_IN * sizeof(float); // 256 KB
  spn_sum_kernel<<<BATCH / ROWS_PER_BLOCK, BLOCK_THREADS, lds_bytes, stream>>>(
      x, pw, idx, out);
}